// SecondOrderGeometryRegularizer_73005854097888
// MI455X (gfx1250) — compile-verified
//
#include <hip/hip_runtime.h>
#include <hip/hip_bf16.h>

// ---------------------------------------------------------------------------
// SecondOrderGeometryRegularizer for MI455X (gfx1250, wave32, WMMA)
//   N=8192 points, D=512 dims, K=32 neighbors, 2048 pairs, scalar output.
// ---------------------------------------------------------------------------

typedef __attribute__((ext_vector_type(2))) float v2f;
typedef __attribute__((ext_vector_type(8))) float v8f;

static constexpr int NPTS   = 8192;
static constexpr int DIM    = 512;
static constexpr int KNEIGH = 32;
static constexpr int KSEL   = 33;   // K+1 (self included)
static constexpr int NPAIRS = 2048;

// ---------------------------------------------------------------------------
// Kernel 1: per-row squared norms.  One wave (32 lanes) per row.
// ---------------------------------------------------------------------------
__global__ __launch_bounds__(256) void sumsq_kernel(const float* __restrict__ E,
                                                    float* __restrict__ sq) {
    const int row  = blockIdx.x * 8 + (threadIdx.x >> 5);
    const int lane = threadIdx.x & 31;
    float s = 0.f;
    #pragma unroll 4
    for (int k = lane; k < DIM; k += 32) {
        float v = E[(size_t)row * DIM + k];
        s += v * v;
    }
    #pragma unroll
    for (int off = 16; off > 0; off >>= 1) s += __shfl_xor(s, off, 32);
    if (lane == 0) sq[row] = s;
}

// ---------------------------------------------------------------------------
// Kernel 2: fused distance GEMM (f32 WMMA 16x16x4) + top-33 selection.
// One workgroup (256 threads = 8 waves) handles a 16-row panel:
//   - A panel (16x512) staged in LDS
//   - outer loop over 8192 cols in tiles of 256 (8 waves x 2 col-tiles of 16)
//   - selection: each wave owns 2 rows; 16 sublanes filter via float4 LDS
//     reads against the running 33rd-smallest threshold; leader drains the
//     (rare) survivors serially.
// ---------------------------------------------------------------------------
__global__ __launch_bounds__(256) void knn_kernel(const float* __restrict__ E,
                                                  const float* __restrict__ sq,
                                                  int* __restrict__ knnOut) {
    __shared__ __align__(16) float sA[16][DIM];     // 32 KB  A panel
    __shared__ __align__(16) float d2p[16][256];    // 16 KB  d^2 tile
    __shared__ float    selVal[16][KSEL];           // selection values
    __shared__ int      selIdx[16][KSEL];           // selection indices
    __shared__ float    selMax[16];                 // running 33rd-smallest
    __shared__ unsigned passMask[16][16];           // per-sublane filter masks
    __shared__ float    sqR[16];                    // row norms

    const int tid     = threadIdx.x;
    const int wave    = tid >> 5;
    const int lane    = tid & 31;
    const int nloc    = lane & 15;     // N / M position within half-wave
    const int khalf   = lane >> 4;     // selects K pair {0,1} vs {2,3}
    const int rowBase = blockIdx.x * 16;
    const int rowSel  = 2 * wave + khalf;   // row this half-wave filters
    const bool leader = (nloc == 0);

    // ---- stage A panel (coalesced float4 loads) ----
    {
        const float4* src = (const float4*)(E + (size_t)rowBase * DIM);
        float4* dst = (float4*)(&sA[0][0]);
        for (int i = tid; i < 16 * DIM / 4; i += 256) dst[i] = src[i];
    }
    if (tid < 16) sqR[tid] = sq[rowBase + tid];
    __syncthreads();

    // leader-held selection state for row rowSel
    int   cnt    = 0;
    float maxVal = -1.0f;
    int   maxPos = 0;

    for (int cb = 0; cb < NPTS; cb += 256) {
        // ---------------- GEMM phase: wave computes two 16x16 tiles ----------
        const int colT0 = cb + wave * 32;            // this wave's 32 columns
        const float* pB0 = E + (size_t)(colT0 + nloc) * DIM;
        const float* pB1 = pB0 + (size_t)16 * DIM;
        const int koff = 2 * khalf;

        v8f acc0 = {};
        v8f acc1 = {};
        #pragma unroll 4
        for (int kk = 0; kk < DIM; kk += 4) {
            const int k0 = kk + koff;
            v2f a  = *(const v2f*)&sA[nloc][k0];     // A: M=nloc, K={k0,k0+1}
            v2f b0 = *(const v2f*)(pB0 + k0);        // B: N=nloc, K={k0,k0+1}
            v2f b1 = *(const v2f*)(pB1 + k0);
            acc0 = __builtin_amdgcn_wmma_f32_16x16x4_f32(
                false, a, false, b0, (short)0, acc0, false, false);
            acc1 = __builtin_amdgcn_wmma_f32_16x16x4_f32(
                false, a, false, b1, (short)0, acc1, false, false);
        }

        // ---------------- epilogue: dot -> clamped squared distance ---------
        const int c0 = wave * 32 + nloc;             // local col of tile 0
        const int c1 = c0 + 16;
        const float sqc0 = sq[cb + c0];
        const float sqc1 = sq[cb + c1];
        #pragma unroll
        for (int v = 0; v < 8; ++v) {
            const int m = v + 8 * khalf;             // C/D layout: M = v (+8 hi)
            const float sr = sqR[m];
            d2p[m][c0] = fmaxf(sr + sqc0 - 2.0f * acc0[v], 0.0f);
            d2p[m][c1] = fmaxf(sr + sqc1 - 2.0f * acc1[v], 0.0f);
        }
        __syncthreads();

        // ---------------- selection phase -----------------------------------
        if (cb == 0) {
            // initial fill: serial insert of first 256 candidates (once)
            if (leader) {
                const int r = rowSel;
                for (int c = 0; c < 256; ++c) {
                    const float d = d2p[r][c];
                    if (cnt < KSEL) {
                        selVal[r][cnt] = d;
                        selIdx[r][cnt] = c;
                        if (d > maxVal) { maxVal = d; maxPos = cnt; }
                        ++cnt;
                    } else if (d < maxVal) {         // ties keep lower index
                        selVal[r][maxPos] = d;
                        selIdx[r][maxPos] = c;
                        maxVal = selVal[r][0]; maxPos = 0;
                        #pragma unroll 4
                        for (int j = 1; j < KSEL; ++j)
                            if (selVal[r][j] > maxVal) { maxVal = selVal[r][j]; maxPos = j; }
                    }
                }
                selMax[r] = maxVal;
            }
        } else {
            const int r = rowSel;
            const float thresh = selMax[r];
            // parallel filter: sublane `nloc` scans cols [nloc*16, nloc*16+16)
            unsigned pm = 0;
            const float4* rowp = (const float4*)&d2p[r][0];
            #pragma unroll
            for (int q = 0; q < 4; ++q) {
                const float4 v = rowp[nloc * 4 + q];
                const int b = q * 4;
                if (v.x < thresh) pm |= 1u << (b + 0);
                if (v.y < thresh) pm |= 1u << (b + 1);
                if (v.z < thresh) pm |= 1u << (b + 2);
                if (v.w < thresh) pm |= 1u << (b + 3);
            }
            passMask[r][nloc] = pm;   // same-wave LDS ops are in-order (DScnt)
            if (leader) {
                for (int t = 0; t < 16; ++t) {
                    unsigned m = passMask[r][t];
                    while (m) {
                        const int c = __builtin_ctz(m); m &= m - 1;
                        const float d = d2p[r][t * 16 + c];
                        if (d < maxVal) {
                            selVal[r][maxPos] = d;
                            selIdx[r][maxPos] = cb + t * 16 + c;
                            maxVal = selVal[r][0]; maxPos = 0;
                            #pragma unroll 4
                            for (int j = 1; j < KSEL; ++j)
                                if (selVal[r][j] > maxVal) { maxVal = selVal[r][j]; maxPos = j; }
                        }
                    }
                }
                selMax[r] = maxVal;
            }
        }
        __syncthreads();
    }

    // ---- finalize: sort 33 by (dist, idx), drop self, emit 32 neighbors ----
    if (tid < 16) {
        const int r = tid;
        for (int i = 1; i < KSEL; ++i) {
            float v = selVal[r][i]; int id = selIdx[r][i];
            int j = i - 1;
            while (j >= 0 && (selVal[r][j] > v ||
                              (selVal[r][j] == v && selIdx[r][j] > id))) {
                selVal[r][j + 1] = selVal[r][j];
                selIdx[r][j + 1] = selIdx[r][j];
                --j;
            }
            selVal[r][j + 1] = v; selIdx[r][j + 1] = id;
        }
        const int self = rowBase + r;
        int out = 0;
        for (int j = 0; j < KSEL && out < KNEIGH; ++j)
            if (selIdx[r][j] != self)
                knnOut[(size_t)self * KNEIGH + (out++)] = selIdx[r][j];
    }
}

// ---------------------------------------------------------------------------
// Kernel 3: pair overlaps. One wave per pair; lane m owns neighbor m.
//   overlap[m] = |knn(a) ∩ knn(knn(a)[m])| / 32; wave rank-sort; MSE partials.
// ---------------------------------------------------------------------------
__global__ __launch_bounds__(256) void pair_kernel(const int* __restrict__ pairIdx,
                                                   const int* __restrict__ knn,
                                                   const float* __restrict__ ref,
                                                   float* __restrict__ partial) {
    __shared__ float sorted[8][KNEIGH];
    __shared__ float wsum[8];
    const int wave = threadIdx.x >> 5;
    const int lane = threadIdx.x & 31;
    const int p    = blockIdx.x * 8 + wave;

    const int a  = pairIdx[p];
    const int aN = knn[(size_t)a * KNEIGH + lane];   // lane's member of set A

    int bl[KNEIGH];
    #pragma unroll
    for (int t = 0; t < KNEIGH; ++t) bl[t] = knn[(size_t)aN * KNEIGH + t];

    int count = 0;
    for (int t = 0; t < KNEIGH; ++t) {
        const int av = __shfl(aN, t, 32);
        int f = 0;
        #pragma unroll
        for (int u = 0; u < KNEIGH; ++u) f |= (bl[u] == av) ? 1 : 0;
        count += f;
    }

    // stable rank sort (overlaps are small ints; ties broken by lane)
    int rank = 0;
    for (int t = 0; t < KNEIGH; ++t) {
        const int ct = __shfl(count, t, 32);
        rank += (ct < count || (ct == count && t < lane)) ? 1 : 0;
    }
    sorted[wave][rank] = (float)count * (1.0f / (float)KNEIGH);
    __syncthreads();

    const float diff = sorted[wave][lane] - ref[(size_t)p * KNEIGH + lane];
    float sqd = diff * diff;
    #pragma unroll
    for (int off = 16; off > 0; off >>= 1) sqd += __shfl_xor(sqd, off, 32);
    if (lane == 0) wsum[wave] = sqd;
    __syncthreads();
    if (threadIdx.x == 0) {
        float s = 0.f;
        #pragma unroll
        for (int i = 0; i < 8; ++i) s += wsum[i];
        partial[blockIdx.x] = s;
    }
}

// ---------------------------------------------------------------------------
// Kernel 4: deterministic tree reduction of 256 partials -> mean loss.
// ---------------------------------------------------------------------------
__global__ __launch_bounds__(256) void reduce_kernel(const float* __restrict__ partial,
                                                     float* __restrict__ out) {
    __shared__ float buf[256];
    const int t = threadIdx.x;
    buf[t] = partial[t];
    __syncthreads();
    #pragma unroll
    for (int off = 128; off > 0; off >>= 1) {
        if (t < off) buf[t] += buf[t + off];
        __syncthreads();
    }
    if (t == 0) out[0] = buf[0] * (1.0f / ((float)NPAIRS * (float)KNEIGH));
}

// ---------------------------------------------------------------------------
extern "C" void kernel_launch(void* const* d_in, const int* in_sizes, int n_in,
                              void* d_out, int out_size, void* d_ws, size_t ws_size,
                              hipStream_t stream) {
    const float* E    = (const float*)d_in[0];   // embeddings [8192,512] f32
    const float* ref  = (const float*)d_in[1];   // reference  [2048,32]  f32
    const int*   pidx = (const int*)d_in[2];     // pair_indices [2048]   i32

    char* ws = (char*)d_ws;
    float* sq      = (float*)ws;                               // 32 KB
    int*   knn     = (int*)(ws + 32768);                       // 1 MB
    float* partial = (float*)(ws + 32768 + (size_t)NPTS * KNEIGH * 4); // 1 KB

    sumsq_kernel <<<NPTS / 8,   256, 0, stream>>>(E, sq);
    knn_kernel   <<<NPTS / 16,  256, 0, stream>>>(E, sq, knn);
    pair_kernel  <<<NPAIRS / 8, 256, 0, stream>>>(pidx, knn, ref, partial);
    reduce_kernel<<<1,          256, 0, stream>>>(partial, (float*)d_out);

    (void)in_sizes; (void)n_in; (void)out_size; (void)ws_size;
}